// cal_context_40183714021651
// MI455X (gfx1250) — compile-verified
//
#include <hip/hip_runtime.h>
#include <math.h>

typedef float v2f __attribute__((ext_vector_type(2)));
typedef float v8f __attribute__((ext_vector_type(8)));

#define B_ 32
#define S_ 4096
#define D_ 256
#define DA_ 128
#define DP_ 64
#define NC_ 16
#define CHUNK_ (S_ / NC_)   // 256 rows of s per block
#define WAVES_ 4
#define NT_ (CHUNK_ / 16 / WAVES_)   // 4 tiles per wave
#define TP_ 260                       // padded LDS tile pitch (floats): 260%64=4 -> conflict-free
#define TILE_F (16 * TP_)             // floats per LDS tile buffer

// workspace layout (in floats)
#define WS_VH   0      // 256
#define WS_VQ   256    // 256
#define WS_VP1  512    // 64
#define WS_VP2  576    // 64
#define WS_QB   640    // 32  (q·vq + bc, per batch)
#define WS_ESUM 672    // B_*NC_ = 512
#define WS_CTXP 1184   // B_*NC_*D_ = 131072

#if defined(__HIP_DEVICE_COMPILE__) && defined(__gfx1250__) && \
    __has_builtin(__builtin_amdgcn_global_load_async_to_lds_b128) && \
    __has_builtin(__builtin_amdgcn_s_wait_asynccnt)
#define USE_ASYNC 1
#else
#define USE_ASYNC 0
#endif

#if USE_ASYNC
typedef int v4i __attribute__((vector_size(16)));
typedef __attribute__((address_space(1))) v4i as1_v4i;   // global int4
typedef __attribute__((address_space(3))) v4i as3_v4i;   // LDS int4
#endif

// ---------------------------------------------------------------------------
// prep1: fold Wc into weights:  vh = Wh^T wa, vq = Wq^T wq, vp* = Wp*^T wp*
__global__ void prep1_kernel(const float* __restrict__ Wh, const float* __restrict__ Wq,
                             const float* __restrict__ Wp1, const float* __restrict__ Wp2,
                             const float* __restrict__ Wc, float* __restrict__ ws) {
    int t = threadIdx.x;
    if (t < 256) {
        float s = 0.f;
        for (int a = 0; a < DA_; ++a) s += Wh[a * D_ + t] * Wc[a];
        ws[WS_VH + t] = s;
    } else if (t < 512) {
        int d = t - 256;
        float s = 0.f;
        for (int a = 0; a < DA_; ++a) s += Wq[a * D_ + d] * Wc[DA_ + a];
        ws[WS_VQ + d] = s;
    } else if (t < 576) {
        int j = t - 512;
        float s = 0.f;
        for (int i = 0; i < DP_; ++i) s += Wp1[i * DP_ + j] * Wc[2 * DA_ + i];
        ws[WS_VP1 + j] = s;
    } else if (t < 640) {
        int j = t - 576;
        float s = 0.f;
        for (int i = 0; i < DP_; ++i) s += Wp2[i * DP_ + j] * Wc[2 * DA_ + DP_ + i];
        ws[WS_VP2 + j] = s;
    }
}

// prep2: qb[b] = q[b,:]·vq + bc
__global__ void prep2_kernel(const float* __restrict__ q, const float* __restrict__ bc,
                             float* __restrict__ ws) {
    int b = threadIdx.x;
    if (b < B_) {
        const float* vq = ws + WS_VQ;
        float s = bc[0];
        for (int d = 0; d < D_; ++d) s += q[b * D_ + d] * vq[d];
        ws[WS_QB + b] = s;
    }
}

// ---------------------------------------------------------------------------
// stage one 16 x 256 f32 tile of h into LDS (pitch TP_), coalesced.
// 32 ops x 32 lanes x 16B = 16 KB.  Async path tracked by ASYNCcnt.
__device__ __forceinline__ void tile_prefetch(const float* __restrict__ g,
                                              float* l, int lid) {
#if USE_ASYNC
#pragma unroll
    for (int r = 0; r < 16; ++r) {
#pragma unroll
        for (int p = 0; p < 2; ++p) {
            const float* gs = g + r * D_ + p * 128 + lid * 4;
            float* ls = l + r * TP_ + p * 128 + lid * 4;
            __builtin_amdgcn_global_load_async_to_lds_b128(
                (as1_v4i*)(void*)const_cast<float*>(gs),
                (as3_v4i*)(void*)ls, 0, 0);
        }
    }
#else
#pragma unroll
    for (int r = 0; r < 16; ++r) {
#pragma unroll
        for (int p = 0; p < 2; ++p) {
            *(float4*)(l + r * TP_ + p * 128 + lid * 4) =
                *(const float4*)(g + r * D_ + p * 128 + lid * 4);
        }
    }
#endif
}

// ---------------------------------------------------------------------------
// main: fused score -> exp(tanh) -> partial weighted sum, single pass over h.
// grid = B_*NC_ blocks, 128 threads (4 waves). Wave processes 16-row s tiles,
// double-buffered through LDS via async global->LDS copies.
__global__ __launch_bounds__(128) void ctx_main_kernel(
    const float* __restrict__ h, const float* __restrict__ pf1,
    const float* __restrict__ pf2, const float* __restrict__ ws,
    float* __restrict__ esum_out, float* __restrict__ ctxp_out) {

    __shared__ __align__(16) float s_tile[WAVES_ * 2 * TILE_F];   // ~133 KB
    __shared__ __align__(16) float s_vh[D_];
    __shared__ __align__(16) float s_vp1[DP_];
    __shared__ __align__(16) float s_vp2[DP_];
    __shared__ __align__(16) float s_score[WAVES_ * 16];
    __shared__ __align__(16) float s_ctxw[WAVES_ * D_];
    __shared__ float s_esw[WAVES_];

    const int tid = threadIdx.x;
    const int bid = blockIdx.x;
    const int b = bid / NC_;
    const int c = bid % NC_;
    const int sbase = c * CHUNK_;

    for (int i = tid; i < D_; i += 128) s_vh[i] = ws[WS_VH + i];
    if (tid < DP_) { s_vp1[tid] = ws[WS_VP1 + tid]; s_vp2[tid] = ws[WS_VP2 + tid]; }
    __syncthreads();

    const float qbb = ws[WS_QB + b];

    const int wid = tid >> 5;
    const int lid = tid & 31;
    const int half = lid >> 4;      // 0: K 0/1 half, 1: K 2/3 half (f32 WMMA A layout)
    const int m = lid & 15;         // row within 16-row tile

    float* buf0 = s_tile + wid * (2 * TILE_F);
    float* buf1 = buf0 + TILE_F;

    float ctx_acc[8];
#pragma unroll
    for (int i = 0; i < 8; ++i) ctx_acc[i] = 0.f;
    float esum_acc = 0.f;

    // tile j (wave-local) covers s0 = sbase + (wid*NT_ + j)*16
    const size_t hrow0 = ((size_t)b * S_ + sbase + wid * NT_ * 16) * D_;

    tile_prefetch(h + hrow0, buf0, lid);

    for (int j = 0; j < NT_; ++j) {
        float* tile = (j & 1) ? buf1 : buf0;
        if (j + 1 < NT_) {
            float* ntile = (j & 1) ? buf0 : buf1;
            tile_prefetch(h + hrow0 + (size_t)(j + 1) * 16 * D_, ntile, lid);
#if USE_ASYNC
            __builtin_amdgcn_s_wait_asynccnt(32);  // tile j resident (in-order)
#endif
        } else {
#if USE_ASYNC
            __builtin_amdgcn_s_wait_asynccnt(0);
#endif
        }

        // ---- score tile (16 s-rows) = h_tile(16xD) . vh  via WMMA f32 16x16x4 ----
        v8f acc = {};
        const float* arow = tile + m * TP_ + 2 * half;   // LDS, conflict-free (pitch 260)
        const float* bcol = s_vh + 2 * half;             // B[k,n] = vh[k] replicated over n
#pragma unroll 8
        for (int k0 = 0; k0 < D_; k0 += 4) {
            v2f av = *(const v2f*)(arow + k0);
            v2f bv = *(const v2f*)(bcol + k0);
            acc = __builtin_amdgcn_wmma_f32_16x16x4_f32(
                false, av, false, bv, (short)0, acc, false, false);
        }
        // column 0 of D: lane 0 holds rows 0..7, lane 16 holds rows 8..15
        if (lid == 0) {
#pragma unroll
            for (int i = 0; i < 8; ++i) s_score[wid * 16 + i] = acc[i];
        }
        if (lid == 16) {
#pragma unroll
            for (int i = 0; i < 8; ++i) s_score[wid * 16 + 8 + i] = acc[i];
        }

        // ---- pf1/pf2 dot products: lane (half,m) covers 32 of 64 dims of row m ----
        const int s0 = sbase + (wid * NT_ + j) * 16;
        float pfacc = 0.f;
        {
            const size_t pbase = ((size_t)b * S_ + s0 + m) * DP_ + half * 32;
            const float* p1 = pf1 + pbase;
            const float* p2 = pf2 + pbase;
            const float* w1 = s_vp1 + half * 32;
            const float* w2 = s_vp2 + half * 32;
#pragma unroll
            for (int jj = 0; jj < 32; jj += 4) {
                float4 x1 = *(const float4*)(p1 + jj);
                float4 y1 = *(const float4*)(w1 + jj);
                float4 x2 = *(const float4*)(p2 + jj);
                float4 y2 = *(const float4*)(w2 + jj);
                pfacc += x1.x * y1.x + x1.y * y1.y + x1.z * y1.z + x1.w * y1.w
                       + x2.x * y2.x + x2.y * y2.y + x2.z * y2.z + x2.w * y2.w;
            }
        }
        pfacc += __shfl_xor(pfacc, 16, 32);

        // ---- e = exp(tanh(score));  tanh-bounded => no softmax max needed ----
        float e = 0.f;
        if (lid < 16) {
            float sc = s_score[wid * 16 + lid] + pfacc + qbb;
            float v = tanhf(sc);
            e = __expf(v);
            esum_acc += e;
        }

        // ---- ctx[d] += e_s * h[s,d]; lane owns d = lid*8 .. lid*8+7 (from LDS) ----
        const float* hr = tile + lid * 8;
#pragma unroll
        for (int s = 0; s < 16; ++s) {
            float es = __shfl(e, s, 32);
            float4 h0 = *(const float4*)(hr + s * TP_);
            float4 h1 = *(const float4*)(hr + s * TP_ + 4);
            ctx_acc[0] += es * h0.x; ctx_acc[1] += es * h0.y;
            ctx_acc[2] += es * h0.z; ctx_acc[3] += es * h0.w;
            ctx_acc[4] += es * h1.x; ctx_acc[5] += es * h1.y;
            ctx_acc[6] += es * h1.z; ctx_acc[7] += es * h1.w;
        }
    }

    // ---- deterministic block reduction (fixed-order, no FP atomics) ----
#pragma unroll
    for (int i = 0; i < 8; ++i) s_ctxw[wid * D_ + lid * 8 + i] = ctx_acc[i];
    esum_acc += __shfl_down(esum_acc, 8, 32);
    esum_acc += __shfl_down(esum_acc, 4, 32);
    esum_acc += __shfl_down(esum_acc, 2, 32);
    esum_acc += __shfl_down(esum_acc, 1, 32);
    if (lid == 0) s_esw[wid] = esum_acc;
    __syncthreads();

    for (int d = tid; d < D_; d += 128) {
        float cx = 0.f;
#pragma unroll
        for (int w = 0; w < WAVES_; ++w) cx += s_ctxw[w * D_ + d];
        ctxp_out[(size_t)bid * D_ + d] = cx;
    }
    if (tid == 0) {
        float es = 0.f;
#pragma unroll
        for (int w = 0; w < WAVES_; ++w) es += s_esw[w];
        esum_out[bid] = es;
    }
}

// ---------------------------------------------------------------------------
// finalize: context[b,d] = (sum_c ctxp) / (sum_c esum) / S
__global__ void finalize_kernel(const float* __restrict__ esum,
                                const float* __restrict__ ctxp,
                                float* __restrict__ out) {
    int b = blockIdx.x;
    int d = threadIdx.x;
    float es = 0.f;
#pragma unroll
    for (int c = 0; c < NC_; ++c) es += esum[b * NC_ + c];
    float cx = 0.f;
#pragma unroll
    for (int c = 0; c < NC_; ++c) cx += ctxp[((size_t)(b * NC_ + c)) * D_ + d];
    out[b * D_ + d] = cx / es * (1.0f / (float)S_);
}

// ---------------------------------------------------------------------------
extern "C" void kernel_launch(void* const* d_in, const int* in_sizes, int n_in,
                              void* d_out, int out_size, void* d_ws, size_t ws_size,
                              hipStream_t stream) {
    (void)in_sizes; (void)n_in; (void)out_size; (void)ws_size;
    const float* h   = (const float*)d_in[0];
    const float* q   = (const float*)d_in[1];
    const float* pf1 = (const float*)d_in[2];
    const float* pf2 = (const float*)d_in[3];
    const float* Wh  = (const float*)d_in[4];
    const float* Wq  = (const float*)d_in[5];
    const float* Wp1 = (const float*)d_in[6];
    const float* Wp2 = (const float*)d_in[7];
    const float* Wc  = (const float*)d_in[8];
    const float* bc  = (const float*)d_in[9];
    float* ws  = (float*)d_ws;
    float* out = (float*)d_out;

    prep1_kernel<<<1, 640, 0, stream>>>(Wh, Wq, Wp1, Wp2, Wc, ws);
    prep2_kernel<<<1, 32, 0, stream>>>(q, bc, ws);
    ctx_main_kernel<<<B_ * NC_, 128, 0, stream>>>(h, pf1, pf2, ws,
                                                  ws + WS_ESUM, ws + WS_CTXP);
    finalize_kernel<<<B_, D_, 0, stream>>>(ws + WS_ESUM, ws + WS_CTXP, out);
}